// FCNet_78898549227817
// MI455X (gfx1250) — compile-verified
//
#include <hip/hip_runtime.h>
#include <hip/hip_bf16.h>

// ---------------------------------------------------------------------------
// MI455X (gfx1250) implementation of the cascaded FCNet + data-consistency
// solver. fp32 throughout (matches reference); GEMMs use V_WMMA_F32_16X16X4_F32
// with async global->LDS staging (ASYNCcnt) and branch-free WMMA loops.
// ---------------------------------------------------------------------------

typedef __attribute__((ext_vector_type(2))) float v2f;
typedef __attribute__((ext_vector_type(8))) float v8f;
typedef __attribute__((address_space(3))) float lds_f;

#define NC 47
#define NFEAT (27 * NC)   // 1269
#define FEAT_LD 1280      // padded leading dim (16B-aligned rows)
#define HID 512
#define DIM_D 300
#define FULL 20

#define BM 32
#define BN 64
#define KC 32
#define ALD 36            // 144B row stride: 16B aligned, conflict-free fragment rows
#define BLD 36

__constant__ int IDX18_c[18] = {0,1,2,3,4,5,6,7,8,9,10,11,12,13,14,15,45,46};

static inline int cdiv_h(int a, int b) { return (a + b - 1) / b; }

// ---- CDNA5 async global->LDS copy (16B per lane), tracked by ASYNCcnt ------
__device__ __forceinline__ void async_copy16(const float* gsrc, float* ldst)
{
    unsigned lds_off = (unsigned)(unsigned long long)(lds_f*)ldst;
    asm volatile("global_load_async_to_lds_b128 %0, %1, off"
                 :: "v"(lds_off), "v"(gsrc) : "memory");
}
__device__ __forceinline__ void wait_async0()
{
#if __has_builtin(__builtin_amdgcn_s_wait_asynccnt)
    __builtin_amdgcn_s_wait_asynccnt(0);
#else
    asm volatile("s_wait_asynccnt 0x0" ::: "memory");
#endif
}

// ---------------------------------------------------------------------------
// Generic fp32 WMMA GEMM:  C[M,N] = A[M,K] * op(B) (+ epilogue)
//   transB=1: B is [N,K] row-major (weights);  transB=0: B is [K,N] row-major.
// 256 threads = 8 waves; wave -> one 16x16 tile; WG -> 32x64; K chunked by 32.
// Full interior chunks stage via global_load_async_to_lds_b128; ragged chunks
// use a guarded zero-fill path. WMMA loop itself is branch-free.
// mode: 0=none, 1=+bias, 2=+bias,BN,ReLU, 3=+bias,ReLU
// ---------------------------------------------------------------------------
__global__ __launch_bounds__(256) void wmma_gemm_kernel(
    const float* __restrict__ A, int lda, long strideA,
    const float* __restrict__ Bm, int ldb, long strideB, int transB,
    float* __restrict__ C, int ldc, long strideC,
    int M, int N, int K, int mode,
    const float* __restrict__ bias, const float* __restrict__ gamma,
    const float* __restrict__ beta, const float* __restrict__ mean,
    const float* __restrict__ var)
{
    __shared__ float Alds[BM * ALD];
    __shared__ float Blds[BN * BLD];

    int batch = blockIdx.z;
    A  += (long)batch * strideA;
    Bm += (long)batch * strideB;
    C  += (long)batch * strideC;

    int tid  = threadIdx.x;
    int wave = tid >> 5;
    int lane = tid & 31;
    int waveM = wave >> 2;          // 0..1
    int waveN = wave & 3;           // 0..3
    int rowBlock = blockIdx.y * BM;
    int colBlock = blockIdx.x * BN;

    int ml = lane & 15;             // M (A-frag) / N (B-frag) within tile
    int kp = (lane >> 4) << 1;      // K pair base: 0 or 2

    int rowA = waveM * 16 + ml;     // row within block tile (A fragment)
    int colB = waveN * 16 + ml;     // col within block tile (B fragment)

    bool fastA_ok = (rowBlock + BM <= M) && ((lda & 3) == 0);
    bool fastB_ok = transB && (colBlock + BN <= N) && ((ldb & 3) == 0);

    v8f acc = {};

    for (int k0 = 0; k0 < K; k0 += KC) {
        bool fullK = (k0 + KC <= K);
        // ---- stage A chunk: BM x KC ----
        if (fullK && fastA_ok) {
            int r = tid >> 3, q = (tid & 7) << 2;        // 32 rows x 8 quads
            async_copy16(A + (long)(rowBlock + r) * lda + k0 + q,
                         &Alds[r * ALD + q]);
        } else {
            for (int idx = tid; idx < BM * KC; idx += 256) {
                int r = idx >> 5, kk = idx & 31;
                int gr = rowBlock + r, gk = k0 + kk;
                Alds[r * ALD + kk] = (gr < M && gk < K) ? A[(long)gr * lda + gk] : 0.0f;
            }
        }
        // ---- stage B chunk: BN x KC, always [col][k] in LDS ----
        if (fullK && fastB_ok) {
            #pragma unroll
            for (int i = 0; i < 2; ++i) {
                int idx = tid + i * 256;
                int c = idx >> 3, q = (idx & 7) << 2;    // 64 cols x 8 quads
                async_copy16(Bm + (long)(colBlock + c) * ldb + k0 + q,
                             &Blds[c * BLD + q]);
            }
        } else {
            for (int idx = tid; idx < BN * KC; idx += 256) {
                int c = idx >> 5, kk = idx & 31;
                int gc = colBlock + c, gk = k0 + kk;
                float v = 0.0f;
                if (gc < N && gk < K)
                    v = transB ? Bm[(long)gc * ldb + gk] : Bm[(long)gk * ldb + gc];
                Blds[c * BLD + kk] = v;
            }
        }
        wait_async0();
        __syncthreads();
        // ---- branch-free WMMA over the chunk ----
        const float* Ap = &Alds[rowA * ALD + kp];
        const float* Bp = &Blds[colB * BLD + kp];
        #pragma unroll
        for (int kk = 0; kk < KC; kk += 4) {
            v2f av = *(const v2f*)(Ap + kk);
            v2f bv = *(const v2f*)(Bp + kk);
            acc = __builtin_amdgcn_wmma_f32_16x16x4_f32(
                false, av, false, bv, (short)0, acc, false, false);
        }
        __syncthreads();
    }

    // ---- epilogue ----
    int col = colBlock + waveN * 16 + ml;
    if (col < N) {
        float bb = (mode >= 1) ? bias[col] : 0.0f;
        float g = 1.0f, bt = 0.0f, mu = 0.0f, rs = 1.0f;
        if (mode == 2) {
            g = gamma[col]; bt = beta[col]; mu = mean[col];
            rs = rsqrtf(var[col] + 1e-5f);
        }
        int rowBase = rowBlock + waveM * 16;
        int mhi = (lane >> 4) * 8;
        #pragma unroll
        for (int r = 0; r < 8; ++r) {
            int row = rowBase + mhi + r;
            if (row < M) {
                float v = acc[r] + bb;
                if (mode == 2) v = (v - mu) * rs * g + bt;
                if (mode >= 2) v = fmaxf(v, 0.0f);
                C[(long)row * ldc + col] = v;
            }
        }
    }
}

// ---------------------------------------------------------------------------
// Repack a [rows x kin] row-major matrix into leading dimension ldo.
// ---------------------------------------------------------------------------
__global__ void padw_kernel(const float* __restrict__ Wsrc, float* __restrict__ Wdst,
                            int rows, int kin, int ldo)
{
    long idx = (long)blockIdx.x * blockDim.x + threadIdx.x;
    long total = (long)rows * kin;
    if (idx >= total) return;
    int r = (int)(idx / kin), k = (int)(idx % kin);
    Wdst[(long)r * ldo + k] = Wsrc[idx];
}

// ---------------------------------------------------------------------------
// AQ_TAQ[b,i,j] = sum_p AQ[b,p,i]*AQ[b,p,j]
// ---------------------------------------------------------------------------
__global__ void aqtaq_kernel(const float* __restrict__ AQ, float* __restrict__ out)
{
    int idx = blockIdx.x * blockDim.x + threadIdx.x;
    if (idx >= 2 * NC * NC) return;
    int b = idx / (NC * NC);
    int e = idx % (NC * NC);
    int i = e / NC, j = e % NC;
    const float* base = AQ + (long)b * DIM_D * NC;
    float s = 0.0f;
    for (int p = 0; p < DIM_D; ++p) s += base[p * NC + i] * base[p * NC + j];
    out[idx] = s;
}

// ---------------------------------------------------------------------------
// A18 = AQ18^T AQ18 + 0.01 I  ->  Gauss-Jordan inverse (one WG per batch)
// ---------------------------------------------------------------------------
__global__ void a18inv_kernel(const float* __restrict__ AQ, float* __restrict__ out)
{
    __shared__ float Mx[18][37];
    __shared__ float colk[18];
    int b = blockIdx.x;
    int tid = threadIdx.x;          // 324 threads
    int i = tid / 18, j = tid % 18;
    {
        const float* base = AQ + (long)b * DIM_D * NC;
        int ii = IDX18_c[i], jj = IDX18_c[j];
        float s = 0.0f;
        for (int p = 0; p < DIM_D; ++p) s += base[p * NC + ii] * base[p * NC + jj];
        Mx[i][j] = s + ((i == j) ? 0.01f : 0.0f);
        Mx[i][18 + j] = (i == j) ? 1.0f : 0.0f;
    }
    __syncthreads();
    for (int k = 0; k < 18; ++k) {
        float piv = Mx[k][k];
        if (tid < 18) colk[tid] = (tid == k) ? 0.0f : Mx[tid][k];
        __syncthreads();
        if (tid < 36) Mx[k][tid] = Mx[k][tid] / piv;
        __syncthreads();
        for (int t = tid; t < 18 * 36; t += 324) {
            int ri = t / 36, cj = t % 36;
            if (ri != k) Mx[ri][cj] -= colk[ri] * Mx[k][cj];
        }
        __syncthreads();
    }
    out[b * 324 + i * 18 + j] = Mx[i][18 + j];
}

// ---------------------------------------------------------------------------
// c_init: c[vox,ch] = scatter(A18inv @ AQ_Tb[vox, IDX18]) at channels 0..15,45,46
// ---------------------------------------------------------------------------
__global__ void c_init_kernel(const float* __restrict__ aqtb,
                              const float* __restrict__ a18inv,
                              float* __restrict__ c)
{
    long idx = (long)blockIdx.x * blockDim.x + threadIdx.x;
    long total = 2L * 8000 * NC;
    if (idx >= total) return;
    int ch = (int)(idx % NC);
    long vox = idx / NC;                 // global voxel (batch folded in)
    int b = (int)(vox / 8000);
    float val = 0.0f;
    int j = -1;
    if (ch < 16) j = ch;
    else if (ch >= 45) j = ch - 29;      // 45->16, 46->17
    if (j >= 0) {
        const float* inv = a18inv + b * 324 + j * 18;
        const float* r = aqtb + vox * NC;
        for (int j2 = 0; j2 < 18; ++j2) val += inv[j2] * r[IDX18_c[j2]];
    }
    c[idx] = val;
}

// ---------------------------------------------------------------------------
// 27-neighborhood gather into padded feature rows (leading dim FEAT_LD)
// ---------------------------------------------------------------------------
__global__ void neigh_kernel(const float* __restrict__ c, float* __restrict__ feats,
                             int dim_in, int dim_out, long total)
{
    long idx = (long)blockIdx.x * blockDim.x + threadIdx.x;
    if (idx >= total) return;
    int f = (int)(idx % NFEAT);
    long vox = idx / NFEAT;
    int z = (int)(vox % dim_out); long t = vox / dim_out;
    int y = (int)(t % dim_out);   t /= dim_out;
    int x = (int)(t % dim_out);   int b = (int)(t / dim_out);
    int ch = f % NC, pat = f / NC;
    int dk = pat % 3, dj = (pat / 3) % 3, di = pat / 9;
    long src = ((((long)(b * dim_in + (x + di)) * dim_in + (y + dj)) * dim_in + (z + dk)) * NC) + ch;
    feats[vox * FEAT_LD + f] = c[src];
}

// w += c[:,1:-1,1:-1,1:-1]
__global__ void addc_kernel(float* __restrict__ w, const float* __restrict__ c,
                            int dim_in, int dim_out, long total)
{
    long idx = (long)blockIdx.x * blockDim.x + threadIdx.x;
    if (idx >= total) return;
    int ch = (int)(idx % NC);
    long vox = idx / NC;
    int z = (int)(vox % dim_out); long t = vox / dim_out;
    int y = (int)(t % dim_out);   t /= dim_out;
    int x = (int)(t % dim_out);   int b = (int)(t / dim_out);
    long src = ((((long)(b * dim_in + (x + 1)) * dim_in + (y + 1)) * dim_in + (z + 1)) * NC) + ch;
    w[idx] += c[src];
}

// rhs = AQ_Tb[:, n:-n, n:-n, n:-n] + lam_d * w
__global__ void rhs_kernel(float* __restrict__ rhs, const float* __restrict__ w,
                           const float* __restrict__ aqtb,
                           const float* __restrict__ lamd_p,
                           int n, int dim_out, long total)
{
    long idx = (long)blockIdx.x * blockDim.x + threadIdx.x;
    if (idx >= total) return;
    float lamd = *lamd_p;
    int ch = (int)(idx % NC);
    long vox = idx / NC;
    int z = (int)(vox % dim_out); long t = vox / dim_out;
    int y = (int)(t % dim_out);   t /= dim_out;
    int x = (int)(t % dim_out);   int b = (int)(t / dim_out);
    long src = ((((long)(b * FULL + (x + n)) * FULL + (y + n)) * FULL + (z + n)) * NC) + ch;
    rhs[idx] = aqtb[src] + lamd * w[idx];
}

// ---------------------------------------------------------------------------
// Fused data-consistency solve, one workgroup per voxel, 3 iterations:
//   sig2 = sigmoid(-alpha * P c)^2
//   A    = AQTAQ + lam_n * P^T diag(sig2) P + lam_d I     (SPD -> no pivoting)
//   c    = A^{-1} rhs   (Gaussian elimination in LDS)
// ---------------------------------------------------------------------------
__global__ __launch_bounds__(256) void dc_solve_kernel(
    const float* __restrict__ c_in, float* __restrict__ c_out,
    const float* __restrict__ rhs, const float* __restrict__ aqtaq,
    const float* __restrict__ P,
    const float* __restrict__ lamn_p, const float* __restrict__ lamd_p,
    const float* __restrict__ alpha_p,
    int dim_in, int dim_out)
{
    extern __shared__ float lds[];
    float* cvec = lds;              // 48
    float* xvec = cvec + 48;        // 48
    float* sig2 = xvec + 48;        // 300
    float* Amat = sig2 + 300;       // 47*48

    int tid = threadIdx.x;
    long vox = blockIdx.x;
    int z = (int)(vox % dim_out); long t = vox / dim_out;
    int y = (int)(t % dim_out);   t /= dim_out;
    int x = (int)(t % dim_out);   int b = (int)(t / dim_out);

    float lamn = *lamn_p, lamd = *lamd_p, alpha = *alpha_p;

    long cin_base = (((long)(b * dim_in + (x + 1)) * dim_in + (y + 1)) * dim_in + (z + 1)) * NC;
    if (tid < NC) cvec[tid] = c_in[cin_base + tid];
    __syncthreads();

    const float* aqtaq_b = aqtaq + (long)b * NC * NC;
    const float* rhs_v = rhs + vox * NC;

    for (int it = 0; it < 3; ++it) {
        // sigmoid(-alpha * P c)^2
        for (int p = tid; p < DIM_D; p += blockDim.x) {
            const float* Pr = P + p * NC;
            float dot = 0.0f;
            for (int i = 0; i < NC; ++i) dot += Pr[i] * cvec[i];
            float s = 1.0f / (1.0f + __expf(alpha * dot));
            sig2[p] = s * s;
        }
        __syncthreads();
        // A = AQTAQ + lam_n * P^T diag(sig2) P + lam_d I
        for (int e = tid; e < NC * NC; e += blockDim.x) {
            int i = e / NC, j = e % NC;
            float acc = 0.0f;
            for (int p = 0; p < DIM_D; ++p)
                acc += sig2[p] * P[p * NC + i] * P[p * NC + j];
            Amat[i * 48 + j] = aqtaq_b[e] + lamn * acc + ((i == j) ? lamd : 0.0f);
        }
        if (tid < NC) xvec[tid] = rhs_v[tid];
        __syncthreads();
        // forward elimination (SPD, no pivoting)
        for (int k = 0; k < NC - 1; ++k) {
            if (tid > k && tid < NC) Amat[tid * 48 + k] /= Amat[k * 48 + k];
            __syncthreads();
            int nr = (NC - 1) - k;      // rows k+1..46
            int nc2 = NC - k;           // cols k+1..46, plus rhs at index nc2-1
            for (int t2 = tid; t2 < nr * nc2; t2 += (int)blockDim.x) {
                int ri = k + 1 + t2 / nc2;
                int cj = t2 % nc2;
                float f = Amat[ri * 48 + k];
                if (cj == nc2 - 1) xvec[ri] -= f * xvec[k];
                else               Amat[ri * 48 + (k + 1 + cj)] -= f * Amat[k * 48 + (k + 1 + cj)];
            }
            __syncthreads();
        }
        // back substitution
        if (tid == 0) {
            for (int k = NC - 1; k >= 0; --k) {
                float s = xvec[k];
                for (int j = k + 1; j < NC; ++j) s -= Amat[k * 48 + j] * xvec[j];
                xvec[k] = s / Amat[k * 48 + k];
            }
        }
        __syncthreads();
        if (tid < NC) cvec[tid] = xvec[tid];
        __syncthreads();
    }
    if (tid < NC) c_out[vox * NC + tid] = cvec[tid];
}

// ---------------------------------------------------------------------------
// Host orchestration
// ---------------------------------------------------------------------------
static void launch_gemm(hipStream_t s, const float* A, int lda, long sA,
                        const float* Bm, int ldb, long sB, int transB,
                        float* C, int ldc, long sC,
                        int M, int N, int K, int batches, int mode,
                        const float* bias, const float* g, const float* bt,
                        const float* mu, const float* var)
{
    dim3 grid(cdiv_h(N, BN), cdiv_h(M, BM), batches);
    wmma_gemm_kernel<<<grid, 256, 0, s>>>(A, lda, sA, Bm, ldb, sB, transB,
                                          C, ldc, sC, M, N, K, mode,
                                          bias, g, bt, mu, var);
}

extern "C" void kernel_launch(void* const* d_in, const int* in_sizes, int n_in,
                              void* d_out, int out_size, void* d_ws, size_t ws_size,
                              hipStream_t stream)
{
    const float* b_in = (const float*)d_in[0];   // (2,20,20,20,300,1)
    const float* AQ   = (const float*)d_in[1];   // (2,300,47)
    const float* P    = (const float*)d_in[2];   // (300,47)
    // params: 4 cascades x [3x(W,b,g,bt,m,v), (W,b), (W,b)] = 22 leaves each
    const float* lamd_p  = (const float*)d_in[n_in - 3];
    const float* lamn_p  = (const float*)d_in[n_in - 2];
    const float* alpha_p = (const float*)d_in[n_in - 1];

    // workspace partition (floats), 256-float aligned
    float* ws = (float*)d_ws;
    size_t off = 0;
    auto alloc = [&](size_t n) { float* p = ws + off; off += (n + 255) & ~(size_t)255; return p; };
    float* aqtb   = alloc(2L * 8000 * NC);
    float* aqtaq  = alloc(2L * NC * NC);
    float* a18inv = alloc(2L * 18 * 18);
    float* cA     = alloc(2L * 8000 * NC);
    float* cB     = alloc(2L * 8000 * NC);
    float* feats  = alloc((size_t)11664 * FEAT_LD);
    float* h1     = alloc((size_t)11664 * HID);
    float* h2     = alloc((size_t)11664 * HID);
    float* wbuf   = alloc((size_t)11664 * NC);
    float* rhsbuf = alloc((size_t)11664 * NC);
    float* w1pad  = alloc((size_t)HID * FEAT_LD);
    (void)ws_size;

    // 1) AQ_TAQ and A18^{-1}
    aqtaq_kernel<<<cdiv_h(2 * NC * NC, 256), 256, 0, stream>>>(AQ, aqtaq);
    a18inv_kernel<<<2, 324, 0, stream>>>(AQ, a18inv);

    // 2) AQ_Tb[b,vox,i] = sum_p b[b,vox,p] * AQ[b,p,i]   (WMMA, per-batch)
    launch_gemm(stream, b_in, DIM_D, 8000L * DIM_D,
                AQ, NC, (long)DIM_D * NC, /*transB=*/0,
                aqtb, NC, 8000L * NC,
                8000, NC, DIM_D, /*batches=*/2, /*mode=*/0,
                nullptr, nullptr, nullptr, nullptr, nullptr);

    // 3) initial c (scatter of 18-channel least-squares solution)
    {
        long total = 2L * 8000 * NC;
        c_init_kernel<<<(int)((total + 255) / 256), 256, 0, stream>>>(aqtb, a18inv, cA);
    }

    // 4) cascades
    float* cur = cA;
    size_t dc_lds = (48 + 48 + 300 + 47 * 48) * sizeof(float);
    for (int n = 1; n <= 4; ++n) {
        int dim_in = FULL - 2 * (n - 1);
        int dim_out = dim_in - 2;
        int V = 2 * dim_out * dim_out * dim_out;
        float* cOut = (n == 4) ? (float*)d_out : ((cur == cA) ? cB : cA);
        const float* const* prm = (const float* const*)(d_in + 3 + (n - 1) * 22);

        // neighborhood features (padded rows)
        {
            long total = (long)V * NFEAT;
            neigh_kernel<<<(int)((total + 255) / 256), 256, 0, stream>>>(cur, feats, dim_in, dim_out, total);
        }
        // repack W1 into 16B-aligned rows for async staging
        {
            long total = (long)HID * NFEAT;
            padw_kernel<<<(int)((total + 255) / 256), 256, 0, stream>>>(prm[0], w1pad, HID, NFEAT, FEAT_LD);
        }
        // MLP: 1269 -> 512 (BN+ReLU) -> 512 (BN+ReLU) -> 512 (BN+ReLU) -> 512 (ReLU) -> 47
        launch_gemm(stream, feats, FEAT_LD, 0, w1pad, FEAT_LD, 0, 1, h1, HID, 0,
                    V, HID, NFEAT, 1, 2, prm[1], prm[2], prm[3], prm[4], prm[5]);
        launch_gemm(stream, h1, HID, 0, prm[6], HID, 0, 1, h2, HID, 0,
                    V, HID, HID, 1, 2, prm[7], prm[8], prm[9], prm[10], prm[11]);
        launch_gemm(stream, h2, HID, 0, prm[12], HID, 0, 1, h1, HID, 0,
                    V, HID, HID, 1, 2, prm[13], prm[14], prm[15], prm[16], prm[17]);
        launch_gemm(stream, h1, HID, 0, prm[18], HID, 0, 1, h2, HID, 0,
                    V, HID, HID, 1, 3, prm[19], nullptr, nullptr, nullptr, nullptr);
        launch_gemm(stream, h2, HID, 0, prm[20], HID, 0, 1, wbuf, NC, 0,
                    V, NC, HID, 1, 1, prm[21], nullptr, nullptr, nullptr, nullptr);

        long totVC = (long)V * NC;
        if (n > 1)
            addc_kernel<<<(int)((totVC + 255) / 256), 256, 0, stream>>>(wbuf, cur, dim_in, dim_out, totVC);
        rhs_kernel<<<(int)((totVC + 255) / 256), 256, 0, stream>>>(rhsbuf, wbuf, aqtb, lamd_p, n, dim_out, totVC);

        dc_solve_kernel<<<V, 256, dc_lds, stream>>>(cur, cOut, rhsbuf, aqtaq, P,
                                                    lamn_p, lamd_p, alpha_p, dim_in, dim_out);
        cur = cOut;
    }
    (void)in_sizes; (void)out_size;
}